// Net_7559142441490
// MI455X (gfx1250) — compile-verified
//
#include <hip/hip_runtime.h>
#include <hip/hip_bf16.h>
#include <math.h>

// MI455X / gfx1250: wave32, WMMA (not MFMA).
typedef __attribute__((ext_vector_type(2))) float v2f;
typedef __attribute__((ext_vector_type(8))) float v8f;

// ---------------------------------------------------------------------------
// Degree / normalization kernels
// ---------------------------------------------------------------------------
__global__ void k_init_deg(float* __restrict__ deg, int n) {
  int i = blockIdx.x * blockDim.x + threadIdx.x;
  if (i < n) deg[i] = 1.0f;  // self-loop contributes 1 to every node's degree
}

__global__ void k_count_deg(const int* __restrict__ dst, float* __restrict__ deg, int E) {
  int e = blockIdx.x * blockDim.x + threadIdx.x;
  if (e < E) atomicAdd(&deg[dst[e]], 1.0f);
}

__global__ void k_rsqrt_inplace(float* __restrict__ v, int n) {
  int i = blockIdx.x * blockDim.x + threadIdx.x;
  if (i < n) v[i] = rsqrtf(v[i]);
}

__global__ void k_edge_norm(const int* __restrict__ src, const int* __restrict__ dst,
                            const float* __restrict__ dinv, float* __restrict__ norm, int E) {
  int e = blockIdx.x * blockDim.x + threadIdx.x;
  if (e < E) norm[e] = dinv[src[e]] * dinv[dst[e]];
}

// ---------------------------------------------------------------------------
// FP32 WMMA GEMM: H[M x N] = A[M x K] * W[K x N]
// One wave owns one 16-row M-tile and NT 16-col N-tiles (covers full N), so A
// is loaded exactly once from memory. Uses V_WMMA_F32_16X16X4_F32.
//
// Per-lane fragment layout (ISA ch 7.12.2):
//   A (16x4 f32, v2f):  M = lane&15,  K = k0 + 2*(lane>>4) + {0,1}
//   B (4x16 f32, v2f):  N = lane&15,  K = k0 + 2*(lane>>4) + {0,1}
//   C/D (16x16, v8f):   vgpr r -> M = r + 8*(lane>>4), N = lane&15
//
// MASK=false: N is a multiple of 16 -> unconditional loads, no EXEC churn.
// MASK=true:  clamp column to N-1 (always in-bounds) + v_cndmask zeroing,
//             keeping loads branch-free (no s_and_saveexec per fragment).
// ---------------------------------------------------------------------------
template <int NT, int WAVES, bool MASK>
__global__ void k_gemm_wmma_f32(const float* __restrict__ A, const float* __restrict__ W,
                                float* __restrict__ H, int M, int K, int N) {
  const int lane = threadIdx.x & 31;
  const int wid  = threadIdx.x >> 5;
  const int half = lane >> 4;   // 0: lanes 0-15, 1: lanes 16-31
  const int l16  = lane & 15;

  const int mtile = blockIdx.x * WAVES + wid;
  const int m0 = mtile * 16;
  if (m0 >= M) return;          // whole-wave early exit; EXEC stays all-ones for WMMA

  v8f c[NT];
#pragma unroll
  for (int t = 0; t < NT; ++t) c[t] = (v8f){0.f, 0.f, 0.f, 0.f, 0.f, 0.f, 0.f, 0.f};

  const float* arow = A + (size_t)(m0 + l16) * K;

#pragma unroll 2
  for (int k0 = 0; k0 < K; k0 += 4) {
    const int ka = k0 + 2 * half;
    v2f a;
    a.x = arow[ka];
    a.y = arow[ka + 1];
    const float* wrow0 = W + (size_t)ka * N;
    const float* wrow1 = W + (size_t)(ka + 1) * N;

    v2f b[NT];
#pragma unroll
    for (int t = 0; t < NT; ++t) {
      const int n = t * 16 + l16;
      const int ncl = MASK ? (n < N ? n : N - 1) : n;  // clamped, always in-bounds
      b[t].x = wrow0[ncl];
      b[t].y = wrow1[ncl];
      if (MASK) {
        const bool ok = (n < N);
        b[t].x = ok ? b[t].x : 0.f;  // v_cndmask, no EXEC manipulation
        b[t].y = ok ? b[t].y : 0.f;
      }
    }
#pragma unroll
    for (int t = 0; t < NT; ++t) {
      c[t] = __builtin_amdgcn_wmma_f32_16x16x4_f32(
          /*neg_a=*/false, a, /*neg_b=*/false, b[t],
          /*c_mod=*/(short)0, c[t], /*reuse_a=*/false, /*reuse_b=*/false);
    }
  }

#pragma unroll
  for (int t = 0; t < NT; ++t) {
    const int n = t * 16 + l16;
    if (!MASK || n < N) {
#pragma unroll
      for (int r = 0; r < 8; ++r) {
        const int m = m0 + r + 8 * half;
        H[(size_t)m * N + n] = c[t][r];
      }
    }
  }
}

// ---------------------------------------------------------------------------
// Edge scatter: agg[dst] += h[src] * norm   (float4 gather + 4 L2 atomics)
// grid.x covers edges, grid.y covers float4 feature chunks (F/4)
// ---------------------------------------------------------------------------
__global__ void k_scatter(const int* __restrict__ src, const int* __restrict__ dst,
                          const float* __restrict__ norm, const float* __restrict__ h,
                          float* __restrict__ agg, int E, int F) {
  const int e = blockIdx.x * blockDim.x + threadIdx.x;
  if (e >= E) return;
  const int fc = blockIdx.y;
  const int s = src[e], d = dst[e];
  const float w = norm[e];
  const size_t sb = (size_t)s * F + fc * 4;
  const size_t db = (size_t)d * F + fc * 4;
  const float4 v = *(const float4*)(h + sb);
  atomicAdd(agg + db + 0, v.x * w);
  atomicAdd(agg + db + 1, v.y * w);
  atomicAdd(agg + db + 2, v.z * w);
  atomicAdd(agg + db + 3, v.w * w);
}

// ---------------------------------------------------------------------------
// Finalize: agg[i,:] += h[i,:] * dinv[i]^2 (self loop) + bias ; optional ReLU
// ---------------------------------------------------------------------------
__global__ void k_finalize(float* __restrict__ agg, const float* __restrict__ h,
                           const float* __restrict__ dinv, const float* __restrict__ bias,
                           int n, int F, int do_relu) {
  const int i = blockIdx.x * blockDim.x + threadIdx.x;
  if (i >= n) return;
  const int fc = blockIdx.y;
  const float di = dinv[i];
  const float w = di * di;
  const size_t base = (size_t)i * F + fc * 4;
  const float4 a  = *(const float4*)(agg + base);
  const float4 hv = *(const float4*)(h + base);
  const float4 b  = *(const float4*)(bias + fc * 4);
  float4 r;
  r.x = a.x + hv.x * w + b.x;
  r.y = a.y + hv.y * w + b.y;
  r.z = a.z + hv.z * w + b.z;
  r.w = a.w + hv.w * w + b.w;
  if (do_relu) {
    r.x = fmaxf(r.x, 0.f); r.y = fmaxf(r.y, 0.f);
    r.z = fmaxf(r.z, 0.f); r.w = fmaxf(r.w, 0.f);
  }
  *(float4*)(agg + base) = r;
}

// ---------------------------------------------------------------------------
// Row-wise log_softmax over F (=40) classes. One wave32 per node; lanes cover
// f = lane and f = lane+32; reductions via __shfl_xor.
// ---------------------------------------------------------------------------
__global__ void k_log_softmax(const float* __restrict__ in, float* __restrict__ out,
                              int n, int F) {
  const int wavesPerBlock = blockDim.x >> 5;
  const int node = blockIdx.x * wavesPerBlock + (threadIdx.x >> 5);
  const int lane = threadIdx.x & 31;
  if (node >= n) return;
  const float* row = in + (size_t)node * F;
  const float v0 = (lane < F)      ? row[lane]      : -INFINITY;
  const float v1 = (lane + 32 < F) ? row[lane + 32] : -INFINITY;
  float m = fmaxf(v0, v1);
#pragma unroll
  for (int off = 16; off > 0; off >>= 1) m = fmaxf(m, __shfl_xor(m, off, 32));
  float s = ((lane < F) ? expf(v0 - m) : 0.f) + ((lane + 32 < F) ? expf(v1 - m) : 0.f);
#pragma unroll
  for (int off = 16; off > 0; off >>= 1) s += __shfl_xor(s, off, 32);
  const float lse = m + logf(s);
  float* orow = out + (size_t)node * F;
  if (lane < F)      orow[lane]      = v0 - lse;
  if (lane + 32 < F) orow[lane + 32] = v1 - lse;
}

// ---------------------------------------------------------------------------
// Orchestration
// ---------------------------------------------------------------------------
extern "C" void kernel_launch(void* const* d_in, const int* in_sizes, int n_in,
                              void* d_out, int out_size, void* d_ws, size_t ws_size,
                              hipStream_t stream) {
  // Inputs (setup_inputs order): x, edge_index, W1, b1, W2, b2, W4, b4
  const float* x   = (const float*)d_in[0];
  const int*   ei  = (const int*)d_in[1];     // [2, E] flattened: row 0 = src, row 1 = dst
  const float* W1  = (const float*)d_in[2];
  const float* b1  = (const float*)d_in[3];
  const float* W2  = (const float*)d_in[4];
  const float* b2  = (const float*)d_in[5];
  const float* W4  = (const float*)d_in[6];
  const float* b4  = (const float*)d_in[7];
  float* out = (float*)d_out;

  const int F0 = 500, F1 = 128, F2 = 64, F3 = 40;
  const int n = in_sizes[0] / F0;     // 100000
  const int E = in_sizes[1] / 2;      // 1600000
  const int* src = ei;
  const int* dst = ei + E;

  // Workspace layout (all offsets 16B aligned): dinv[n], norm[E], h[n*128], agg[n*128]
  float* dinv = (float*)d_ws;
  float* norm = dinv + n;
  float* h    = norm + E;
  float* agg  = h + (size_t)n * F1;

  const int T = 256;
  const int nB = (n + T - 1) / T;
  const int eB = (E + T - 1) / T;

  // --- degrees & edge norms (once; reused by all 3 layers) ---
  k_init_deg<<<nB, T, 0, stream>>>(dinv, n);
  k_count_deg<<<eB, T, 0, stream>>>(dst, dinv, E);
  k_rsqrt_inplace<<<nB, T, 0, stream>>>(dinv, n);
  k_edge_norm<<<eB, T, 0, stream>>>(src, dst, dinv, norm, E);

  const int WAVES = 8;                       // 8 wave32 per block
  const int mtiles = (n + 15) / 16;
  const int gemmB = (mtiles + WAVES - 1) / WAVES;

  // ---------------- Layer 1: 500 -> 128, ReLU ----------------
  k_gemm_wmma_f32<8, WAVES, false><<<gemmB, WAVES * 32, 0, stream>>>(x, W1, h, n, F0, F1);
  hipMemsetAsync(agg, 0, (size_t)n * F1 * sizeof(float), stream);
  k_scatter<<<dim3(eB, F1 / 4), T, 0, stream>>>(src, dst, norm, h, agg, E, F1);
  k_finalize<<<dim3(nB, F1 / 4), T, 0, stream>>>(agg, h, dinv, b1, n, F1, 1);

  // ---------------- Layer 2: 128 -> 64, ReLU -----------------
  k_gemm_wmma_f32<4, WAVES, false><<<gemmB, WAVES * 32, 0, stream>>>(agg, W2, h, n, F1, F2);
  hipMemsetAsync(agg, 0, (size_t)n * F2 * sizeof(float), stream);
  k_scatter<<<dim3(eB, F2 / 4), T, 0, stream>>>(src, dst, norm, h, agg, E, F2);
  k_finalize<<<dim3(nB, F2 / 4), T, 0, stream>>>(agg, h, dinv, b2, n, F2, 1);

  // ---------------- Layer 3: 64 -> 40, log_softmax -----------
  k_gemm_wmma_f32<3, WAVES, true><<<gemmB, WAVES * 32, 0, stream>>>(agg, W4, h, n, F2, F3);
  hipMemsetAsync(agg, 0, (size_t)n * F3 * sizeof(float), stream);
  k_scatter<<<dim3(eB, F3 / 4), T, 0, stream>>>(src, dst, norm, h, agg, E, F3);
  k_finalize<<<dim3(nB, F3 / 4), T, 0, stream>>>(agg, h, dinv, b4, n, F3, 0);

  const int lsWaves = 8;
  const int lsB = (n + lsWaves - 1) / lsWaves;
  k_log_softmax<<<lsB, lsWaves * 32, 0, stream>>>(agg, out, n, F3);
}